// LuongAttention_26955214750043
// MI455X (gfx1250) — compile-verified
//
#include <hip/hip_runtime.h>
#include <hip/hip_bf16.h>
#include <math.h>

#define S_LEN 4096
#define B_N   64
#define D_N   256

#define NC    8                 // s-chunks per batch row
#define WAVES 4                 // waves per block in attn_partial
#define NP    (NC * WAVES)      // partials per batch row = 32
#define SPW   (S_LEN / NP)      // s-values per wave = 128
#define PART_STRIDE 258         // m, l, c[256]

typedef __attribute__((ext_vector_type(2))) float v2f;
typedef __attribute__((ext_vector_type(4))) float v4f;
typedef __attribute__((ext_vector_type(8))) float v8f;

// ---------------------------------------------------------------------------
// Kernel 1: key = query @ W^T + bias   (M=B=64, N=D=256, K=D=256)
// One wave per 16x16 output tile; V_WMMA_F32_16X16X4_F32, K-loop step 4.
// A[m,k] = query[btile+m, k0+k];  B[k,n] = W[ntile+n, k0+k]  (W transposed).
// Lane layout (ISA 7.12.2): lanes 0-15 hold K={0,1}, lanes 16-31 hold K={2,3}.
// ---------------------------------------------------------------------------
__global__ void key_gemm_wmma(const float* __restrict__ query,
                              const float* __restrict__ W,
                              const float* __restrict__ bias,
                              float* __restrict__ key)
{
    const int lane  = threadIdx.x;          // 0..31 (one wave per block)
    const int btile = blockIdx.x * 16;      // B dimension tile
    const int ntile = blockIdx.y * 16;      // D dimension tile
    const int l15   = lane & 15;
    const int koff  = (lane < 16) ? 0 : 2;

    const float* arow = query + (size_t)(btile + l15) * D_N + koff;
    const float* brow = W     + (size_t)(ntile + l15) * D_N + koff;

    v8f acc = {};
    for (int k0 = 0; k0 < D_N; k0 += 4) {
        v2f a = *(const v2f*)(arow + k0);   // A: K=koff, koff+1
        v2f bm = *(const v2f*)(brow + k0);  // B: K=koff, koff+1 (row of W = col of W^T)
        acc = __builtin_amdgcn_wmma_f32_16x16x4_f32(
            /*neg_a=*/false, a, /*neg_b=*/false, bm,
            /*c_mod=*/(short)0, acc, /*reuse_a=*/false, /*reuse_b=*/false);
    }

    // C/D layout: lane(&15) = N column; VGPR r -> M = r (+8 for lanes 16-31).
    const int   n     = ntile + l15;
    const int   mbase = btile + ((lane < 16) ? 0 : 8);
    const float bs    = bias[n];
    #pragma unroll
    for (int r = 0; r < 8; ++r)
        key[(size_t)(mbase + r) * D_N + n] = acc[r] + bs;
}

// ---------------------------------------------------------------------------
// Kernel 2: online-softmax partials. One wave owns SPW consecutive s rows of
// one batch row b. Lane t owns d in [8t, 8t+8). Single read of `attended`,
// masked tail skipped entirely. No intra-block synchronization.
// partials layout: part[b][p][0]=m, [1]=l, [2..257]=c[256]
// ---------------------------------------------------------------------------
__global__ void attn_partial(const float* __restrict__ attended,
                             const int*   __restrict__ seqlen,
                             const float* __restrict__ key,
                             float*       __restrict__ part)
{
    const int b     = blockIdx.x;
    const int chunk = blockIdx.y;
    const int wave  = threadIdx.x >> 5;
    const int lane  = threadIdx.x & 31;
    const int p     = chunk * WAVES + wave;
    const int s0    = p * SPW;
    const int len   = seqlen[b];

    // Preload this lane's key segment (8 floats).
    float kreg[8];
    {
        const float* kp = key + (size_t)b * D_N + lane * 8;
        v4f k0 = *(const v4f*)(kp);
        v4f k1 = *(const v4f*)(kp + 4);
        kreg[0]=k0.x; kreg[1]=k0.y; kreg[2]=k0.z; kreg[3]=k0.w;
        kreg[4]=k1.x; kreg[5]=k1.y; kreg[6]=k1.z; kreg[7]=k1.w;
    }

    float m = -INFINITY;
    float l = 0.f;
    float c[8];
    #pragma unroll
    for (int j = 0; j < 8; ++j) c[j] = 0.f;

    const size_t rowStride = (size_t)B_N * D_N;       // floats between s rows
    const float* row = attended + ((size_t)s0 * B_N + b) * D_N + lane * 8;

    const int sEnd = (s0 + SPW < len) ? (s0 + SPW) : len;   // skip masked tail
    for (int s = s0; s < sEnd; ++s, row += rowStride) {
        __builtin_prefetch((const void*)(row + 2 * rowStride), 0, 0);

        v4f a0 = *(const v4f*)(row);
        v4f a1 = *(const v4f*)(row + 4);
        float a[8] = {a0.x, a0.y, a0.z, a0.w, a1.x, a1.y, a1.z, a1.w};

        // Per-lane partial dot, then wave32 reduction -> full 256-dot score.
        float d = 0.f;
        #pragma unroll
        for (int j = 0; j < 8; ++j) d = fmaf(a[j], kreg[j], d);
        #pragma unroll
        for (int off = 16; off >= 1; off >>= 1)
            d += __shfl_xor(d, off, 32);
        const float score = d;

        // Online softmax update.
        const float mnew  = fmaxf(m, score);
        const float scale = __expf(m - mnew);        // 0 on first iter (m=-inf)
        const float e     = __expf(score - mnew);
        l = l * scale + e;
        #pragma unroll
        for (int j = 0; j < 8; ++j) c[j] = fmaf(c[j], scale, e * a[j]);
        m = mnew;
    }

    float* out = part + ((size_t)b * NP + p) * PART_STRIDE;
    if (lane == 0) { out[0] = m; out[1] = l; }
    #pragma unroll
    for (int j = 0; j < 8; ++j) out[2 + lane * 8 + j] = c[j];
}

// ---------------------------------------------------------------------------
// Kernel 3: merge NP partials per batch row; out[b,d] = sum_p e^{m_p-M} c_p[d] / L
// ---------------------------------------------------------------------------
__global__ void attn_combine(const float* __restrict__ part,
                             float*       __restrict__ out)
{
    const int b = blockIdx.x;
    const int d = threadIdx.x;                  // 0..255
    const float* pb = part + (size_t)b * NP * PART_STRIDE;

    float M = -INFINITY;
    #pragma unroll 4
    for (int p = 0; p < NP; ++p)
        M = fmaxf(M, pb[(size_t)p * PART_STRIDE]);

    float L = 0.f, acc = 0.f;
    for (int p = 0; p < NP; ++p) {
        const float* pp = pb + (size_t)p * PART_STRIDE;
        const float  w  = __expf(pp[0] - M);    // 0 for empty (m=-inf) partials
        L   += pp[1] * w;
        acc += pp[2 + d] * w;
    }
    out[(size_t)b * D_N + d] = acc / L;
}

// ---------------------------------------------------------------------------
extern "C" void kernel_launch(void* const* d_in, const int* in_sizes, int n_in,
                              void* d_out, int out_size, void* d_ws, size_t ws_size,
                              hipStream_t stream)
{
    (void)in_sizes; (void)n_in; (void)out_size; (void)ws_size;

    const float* query    = (const float*)d_in[0];   // [B, D]
    const float* attended = (const float*)d_in[1];   // [S, B, D]
    const int*   seqlen   = (const int*)  d_in[2];   // [B]
    const float* W        = (const float*)d_in[3];   // [D, D]
    const float* bias     = (const float*)d_in[4];   // [D]
    float*       out      = (float*)d_out;           // [B, D]

    float* key  = (float*)d_ws;                      // B*D floats = 64 KB
    float* part = key + (size_t)B_N * D_N;           // B*NP*258 floats ~ 2.1 MB

    key_gemm_wmma<<<dim3(B_N / 16, D_N / 16), 32, 0, stream>>>(query, W, bias, key);
    attn_partial <<<dim3(B_N, NC), WAVES * 32, 0, stream>>>(attended, seqlen, key, part);
    attn_combine <<<B_N, D_N, 0, stream>>>(part, out);
}